// FilterPart_37795712205047
// MI455X (gfx1250) — compile-verified
//
#include <hip/hip_runtime.h>
#include <hip/hip_bf16.h>

// ---------------------------------------------------------------------------
// FilterPart query=True path on MI455X (gfx1250):
//   scores[7,N] = emb[7,2049] @ input[N,2049]^T ; clip with out2[N]; max over N.
// Memory-bound (~3.5 FLOP/byte, 656 MB streamed once -> ~28us @ 23.3 TB/s).
// Strategy: stream input once via async global->LDS B128 copies (ASYNCcnt
// double buffering), do the K=7 dot products with V_WMMA_F32_16X16X4_F32
// (emb padded to 16 M-rows, junk rows discarded), fused min/max epilogue.
// This revision: dual WMMA accumulators (halve the C RAW chain) and 8 waves
// per block with 64-wide slabs -> ~124 KB LDS, 2 WG/WGP, 4 waves/SIMD.
// ---------------------------------------------------------------------------

typedef __attribute__((ext_vector_type(2))) float v2f;
typedef __attribute__((ext_vector_type(8))) float v8f;

#define DDIM 2049
#define KROWS 7
#define EMB_STRIDE 2056            // 2056 % 64 == 8 -> emb rows hit distinct banks
#define TILE_D 64
#define BUF_STRIDE 68              // 68 % 64 == 4 -> input rows hit distinct banks
#define NCHUNK 32                  // 32*64 = 2048 ; column d=2048 handled as tail
#define WAVES_PER_BLOCK 8
#define OPS_PER_CHUNK 8            // 16 rows x 256 B/row ; 512 B per async b128 op
#define EMB_FLOATS (KROWS * EMB_STRIDE)
#define BUF_FLOATS (16 * BUF_STRIDE)
#define WAVE_BUF_FLOATS (2 * BUF_FLOATS)
#define SMEM_FLOATS (EMB_FLOATS + WAVES_PER_BLOCK * WAVE_BUF_FLOATS)

// 8 async-to-LDS B128 ops: one 16-row x 64-col fp32 tile (4 KB).
// Op i covers rows 2i,2i+1: lane l -> row 2i+(l>>4), cols (l&15)*4 .. +3.
__device__ __forceinline__ void prefetch_chunk(const float* __restrict__ in,
                                               int n0, int d0, unsigned ldsBase,
                                               int lane) {
  const int rsub = lane >> 4;          // 0/1: which row of the pair
  const int csub = (lane & 15) * 4;    // float column within the row
  // ensure prior ds reads of this (recycled) buffer have drained
  asm volatile("s_wait_dscnt 0" ::: "memory");
#pragma unroll
  for (int i = 0; i < OPS_PER_CHUNK; ++i) {
    const int r = 2 * i + rsub;
    const float* gp = in + (size_t)(n0 + r) * DDIM + d0 + csub;
    unsigned lds = ldsBase + (unsigned)((r * BUF_STRIDE + csub) * sizeof(float));
    asm volatile("global_load_async_to_lds_b128 %0, %1, off"
                 :: "v"(lds), "v"(gp)
                 : "memory");
  }
}

__global__ void __launch_bounds__(WAVES_PER_BLOCK * 32)
filterpart_gemv7(const float* __restrict__ input, const float* __restrict__ out2,
                 const float* __restrict__ emb, float* __restrict__ outMat,
                 float* __restrict__ partials, int nTiles, int N) {
  extern __shared__ float smem[];
  float* embS = smem;
  const int tid  = threadIdx.x;
  const int lane = tid & 31;
  const int wave = tid >> 5;

  // Stage emb (7 x 2049) into LDS, zero the pad columns (L2-resident source).
  for (int idx = tid; idx < EMB_FLOATS; idx += blockDim.x) {
    int r = idx / EMB_STRIDE;
    int col = idx - r * EMB_STRIDE;
    embS[idx] = (col < DDIM) ? emb[r * DDIM + col] : 0.0f;
  }
  __syncthreads();

  float* wbuf = smem + EMB_FLOATS + wave * WAVE_BUF_FLOATS;
  const unsigned wbufByte0 = (unsigned)(size_t)wbuf;
  const unsigned wbufByte1 = (unsigned)(size_t)(wbuf + BUF_FLOATS);

  const int tile = blockIdx.x * WAVES_PER_BLOCK + wave;  // one 16-row tile / wave
  const int n0 = tile * 16;

  const int mrow  = lane & 15;
  const int rowA  = (mrow < KROWS) ? mrow : (KROWS - 1);  // M>=7: dup row -> junk rows, discarded
  const int khalf = (lane >> 4) << 1;                     // 0 for lanes 0-15, 2 for 16-31
  const float* aBase  = embS + rowA * EMB_STRIDE + khalf; // A 16x4: K=k,k+1 | k+2,k+3
  const float* bBase0 = wbuf + mrow * BUF_STRIDE + khalf; // B 4x16: same (K,K+1) pattern
  const float* bBase1 = bBase0 + BUF_FLOATS;

  v8f c0 = {};
  v8f c1 = {};

  prefetch_chunk(input, n0, 0, wbufByte0, lane);
  prefetch_chunk(input, n0, TILE_D, wbufByte1, lane);

  for (int ci = 0; ci < NCHUNK; ++ci) {
    if (ci + 1 < NCHUNK)
      asm volatile("s_wait_asynccnt 8" ::: "memory");   // oldest chunk (8 ops) done
    else
      asm volatile("s_wait_asynccnt 0" ::: "memory");   // final chunk done
    const float* aRow = aBase + ci * TILE_D;
    const float* bRow = (ci & 1) ? bBase1 : bBase0;
    // 16 WMMAs per slab, alternating accumulators to break the C RAW chain.
#pragma unroll
    for (int s = 0; s < TILE_D / 4; s += 2) {
      v2f a0 = *(const v2f*)(aRow + s * 4);
      v2f b0 = *(const v2f*)(bRow + s * 4);
      c0 = __builtin_amdgcn_wmma_f32_16x16x4_f32(false, a0, false, b0, (short)0,
                                                 c0, false, false);
      v2f a1 = *(const v2f*)(aRow + s * 4 + 4);
      v2f b1 = *(const v2f*)(bRow + s * 4 + 4);
      c1 = __builtin_amdgcn_wmma_f32_16x16x4_f32(false, a1, false, b1, (short)0,
                                                 c1, false, false);
    }
    if (ci + 2 < NCHUNK)
      prefetch_chunk(input, n0, (ci + 2) * TILE_D,
                     (ci & 1) ? wbufByte1 : wbufByte0, lane);
  }

  v8f c = c0 + c1;   // fragments share the per-lane C/D layout -> legal add

  // Tail column d = 2048 + fused min/max epilogue.
  // C layout: VGPR v, lanes 0-15 hold (M=v, n=n0+lane); rows 7..15 are junk.
  float xin = 0.0f, o2v = 0.0f;
  if (lane < 16) {
    xin = input[(size_t)(n0 + lane) * DDIM + (DDIM - 1)];
    o2v = out2[n0 + lane];
  }
#pragma unroll
  for (int v = 0; v < KROWS; ++v) {
    float val = c[v];
    if (lane < 16) val += embS[v * EMB_STRIDE + (DDIM - 1)] * xin;
    val = fminf(val, o2v);
    if (lane < 16) outMat[(size_t)v * N + n0 + lane] = val;
    float m = (lane < 16) ? val : -__builtin_huge_valf();
#pragma unroll
    for (int off = 8; off >= 1; off >>= 1)
      m = fmaxf(m, __shfl_xor(m, off, 32));   // xor<16 keeps halves separate
    if (lane == 0) partials[v * nTiles + tile] = m;
  }
}

__global__ void reduce7(const float* __restrict__ partials, float* __restrict__ out,
                        int nTiles) {
  __shared__ float red[256];
  const int tid = threadIdx.x;
  for (int v = 0; v < KROWS; ++v) {
    float m = -__builtin_huge_valf();
    for (int i = tid; i < nTiles; i += 256)
      m = fmaxf(m, partials[v * nTiles + i]);
    red[tid] = m;
    __syncthreads();
    for (int off = 128; off > 0; off >>= 1) {
      if (tid < off) red[tid] = fmaxf(red[tid], red[tid + off]);
      __syncthreads();
    }
    if (tid == 0) out[v] = red[0];
    __syncthreads();
  }
}

extern "C" void kernel_launch(void* const* d_in, const int* in_sizes, int n_in,
                              void* d_out, int out_size, void* d_ws, size_t ws_size,
                              hipStream_t stream) {
  const float* input = (const float*)d_in[0];  // [N, 2049]
  const float* out2  = (const float*)d_in[1];  // [N]
  const float* emb   = (const float*)d_in[2];  // [7, 2049]
  float* out = (float*)d_out;                  // [7] output ++ [7,N] output2
  float* ws  = (float*)d_ws;                   // 7 * nTiles partial maxima

  const int N = in_sizes[1];                   // 80000 (divisible by 128)
  const int nTiles = N / 16;                   // 5000
  const int blocks = nTiles / WAVES_PER_BLOCK; // 625
  const size_t smemBytes = (size_t)SMEM_FLOATS * sizeof(float); // ~124 KB dyn LDS

  filterpart_gemv7<<<blocks, WAVES_PER_BLOCK * 32, smemBytes, stream>>>(
      input, out2, emb, out + KROWS, ws, nTiles, N);
  reduce7<<<1, 256, 0, stream>>>(ws, out, nTiles);
}